// GSGNet_17076789969651
// MI455X (gfx1250) — compile-verified
//
#include <hip/hip_runtime.h>
#include <math.h>

typedef __attribute__((ext_vector_type(2))) float v2f;
typedef __attribute__((ext_vector_type(8))) float v8f;

#define N_NODES 200000
#define N_EDGES 6400000
#define F_IN    16
#define F_H     40
#define F_OUT   24
#define MT      (N_NODES / 16)   // 12500 row tiles

// ---------------------------------------------------------------------------
// inv_deg[i] = deg>0 ? 1/deg : 0   (in place)
// ---------------------------------------------------------------------------
__global__ void invdeg_kernel(float* __restrict__ deg) {
    int i = blockIdx.x * blockDim.x + threadIdx.x;
    if (i >= N_NODES) return;
    float d = deg[i];
    deg[i] = (d > 0.0f) ? (1.0f / d) : 0.0f;
}

// ---------------------------------------------------------------------------
// Layer-1 aggregation fused with degree count:
//   agg1[tgt] += x[src] (16 floats, float4 chunks); deg[tgt] += 1 (chunk 0)
// ---------------------------------------------------------------------------
__global__ void agg1_kernel(const float* __restrict__ x,
                            const long long* __restrict__ src,
                            const long long* __restrict__ tgt,
                            float* __restrict__ agg,
                            float* __restrict__ deg) {
    int tid = blockIdx.x * blockDim.x + threadIdx.x;
    if (tid >= N_EDGES * 4) return;
    int e = tid >> 2;
    int c = tid & 3;
    int s = (int)src[e];
    int t = (int)tgt[e];
    if (c == 0) atomicAdd(&deg[t], 1.0f);
    const float4 v = *(const float4*)(x + (size_t)s * F_IN + c * 4);
    float* dst = agg + (size_t)t * F_IN + c * 4;
    atomicAdd(dst + 0, v.x);
    atomicAdd(dst + 1, v.y);
    atomicAdd(dst + 2, v.z);
    atomicAdd(dst + 3, v.w);
}

// ---------------------------------------------------------------------------
// Layer-2 aggregation: agg2[tgt] += p2[src]  (24 floats, 6 float4 chunks)
// ---------------------------------------------------------------------------
__global__ void agg2_kernel(const float* __restrict__ p2,
                            const long long* __restrict__ src,
                            const long long* __restrict__ tgt,
                            float* __restrict__ agg) {
    int tid = blockIdx.x * blockDim.x + threadIdx.x;
    if (tid >= N_EDGES * 6) return;
    int e = tid / 6;
    int c = tid % 6;
    int s = (int)src[e];
    int t = (int)tgt[e];
    const float4 v = *(const float4*)(p2 + (size_t)s * F_OUT + c * 4);
    float* dst = agg + (size_t)t * F_OUT + c * 4;
    atomicAdd(dst + 0, v.x);
    atomicAdd(dst + 1, v.y);
    atomicAdd(dst + 2, v.z);
    atomicAdd(dst + 3, v.w);
}

// ---------------------------------------------------------------------------
// Layer 1: h = relu( (agg1*inv_deg) @ W1l + x @ W1r + b1 )
// Weights staged zero-padded into LDS (n-major, padded stride 18 so each
// (kb,kb+1) pair is one aligned ds_load_b64). One wave per 16x16 tile,
// V_WMMA_F32_16X16X4_F32, K=16, both matmuls chained into one accumulator.
// ---------------------------------------------------------------------------
__global__ void layer1_kernel(const float* __restrict__ agg,
                              const float* __restrict__ x,
                              const float* __restrict__ invdeg,
                              const float* __restrict__ W1l,
                              const float* __restrict__ b1,
                              const float* __restrict__ W1r,
                              float* __restrict__ h) {
    const int NT = 3;                  // 40 cols -> 3 tiles of 16
    const int NP = 48;                 // padded N
    const int KP = 18;                 // padded K stride (even -> 8B aligned)
    __shared__ float sWl[NP][KP];
    __shared__ float sWr[NP][KP];
    __shared__ float sb[NP];

    // cooperative zero-padded transpose load of W1l/W1r: [K=16][N=40] -> [n][k]
    for (int i = threadIdx.x; i < NP * 16; i += 256) {
        int n = i >> 4, k = i & 15;
        float wl = 0.0f, wr = 0.0f;
        if (n < F_H) {
            wl = W1l[(size_t)k * F_H + n];
            wr = W1r[(size_t)k * F_H + n];
        }
        sWl[n][k] = wl;
        sWr[n][k] = wr;
    }
    if (threadIdx.x < NP) sb[threadIdx.x] = (threadIdx.x < F_H) ? b1[threadIdx.x] : 0.0f;
    __syncthreads();

    int wave = (blockIdx.x * blockDim.x + threadIdx.x) >> 5;
    int lane = threadIdx.x & 31;
    if (wave >= MT * NT) return;
    int mt = wave / NT, nt = wave % NT;
    int row0 = mt * 16, n0 = nt * 16;

    int half = lane >> 4;              // 0: K {0,1}, M 0..7 ; 1: K {2,3}, M 8..15
    int m    = lane & 15;
    int row  = row0 + m;
    int n    = n0 + m;
    float inv = invdeg[row];

    v8f acc = {};
    #pragma unroll
    for (int kk = 0; kk < F_IN; kk += 4) {
        int kb = kk + half * 2;        // even -> 8-byte aligned pair loads
        v2f aAgg = *(const v2f*)(agg + (size_t)row * F_IN + kb);
        v2f aX   = *(const v2f*)(x   + (size_t)row * F_IN + kb);
        aAgg.x *= inv;
        aAgg.y *= inv;
        v2f bL = *(const v2f*)(&sWl[n][kb]);
        v2f bR = *(const v2f*)(&sWr[n][kb]);
        acc = __builtin_amdgcn_wmma_f32_16x16x4_f32(false, aAgg, false, bL,
                                                    (short)0, acc, false, false);
        acc = __builtin_amdgcn_wmma_f32_16x16x4_f32(false, aX, false, bR,
                                                    (short)0, acc, false, false);
    }

    if (n < F_H) {
        float bias = sb[n];
        #pragma unroll
        for (int i = 0; i < 8; ++i) {
            int mm = half * 8 + i;     // C/D layout: lane half selects M vs M+8
            float v = acc[i] + bias;
            h[(size_t)(row0 + mm) * F_H + n] = v > 0.0f ? v : 0.0f;
        }
    }
}

// ---------------------------------------------------------------------------
// Shared layer-2 GEMM body: acc = h_tile @ W (W zero-padded in LDS [32][42])
// ---------------------------------------------------------------------------
__device__ __forceinline__ void stage_w2(const float* __restrict__ W, float (*sW)[42]) {
    for (int i = threadIdx.x; i < 32 * F_H; i += 256) {
        int n = i / F_H, k = i % F_H;
        sW[n][k] = (n < F_OUT) ? W[(size_t)k * F_OUT + n] : 0.0f;
    }
}

__device__ __forceinline__ v8f gemm2_tile(const float* __restrict__ h,
                                          const float (*sW)[42],
                                          int row, int n, int half) {
    v8f acc = {};
    #pragma unroll
    for (int kk = 0; kk < F_H; kk += 4) {
        int kb = kk + half * 2;
        v2f a = *(const v2f*)(h + (size_t)row * F_H + kb);
        v2f b = *(const v2f*)(&sW[n][kb]);
        acc = __builtin_amdgcn_wmma_f32_16x16x4_f32(false, a, false, b,
                                                    (short)0, acc, false, false);
    }
    return acc;
}

// ---------------------------------------------------------------------------
// p2 = h @ W2l  (200k x 40 @ 40 x 24)
// ---------------------------------------------------------------------------
__global__ void gemm_p2_kernel(const float* __restrict__ h,
                               const float* __restrict__ W2l,
                               float* __restrict__ p2) {
    const int NT = 2;
    __shared__ float sW[32][42];
    stage_w2(W2l, sW);
    __syncthreads();

    int wave = (blockIdx.x * blockDim.x + threadIdx.x) >> 5;
    int lane = threadIdx.x & 31;
    if (wave >= MT * NT) return;
    int mt = wave / NT, nt = wave % NT;
    int row0 = mt * 16, n0 = nt * 16;
    int half = lane >> 4, m = lane & 15;
    int row = row0 + m, n = n0 + m;

    v8f acc = gemm2_tile(h, sW, row, n, half);

    if (n < F_OUT) {
        #pragma unroll
        for (int i = 0; i < 8; ++i) {
            int mm = half * 8 + i;
            p2[(size_t)(row0 + mm) * F_OUT + n] = acc[i];
        }
    }
}

// ---------------------------------------------------------------------------
// out = agg2*inv_deg + h @ W2r + b2   (pre-softmax, to d_out)
// ---------------------------------------------------------------------------
__global__ void layer2_out_kernel(const float* __restrict__ h,
                                  const float* __restrict__ W2r,
                                  const float* __restrict__ agg2,
                                  const float* __restrict__ invdeg,
                                  const float* __restrict__ b2,
                                  float* __restrict__ out) {
    const int NT = 2;
    __shared__ float sW[32][42];
    __shared__ float sb[32];
    stage_w2(W2r, sW);
    if (threadIdx.x < 32) sb[threadIdx.x] = (threadIdx.x < F_OUT) ? b2[threadIdx.x] : 0.0f;
    __syncthreads();

    int wave = (blockIdx.x * blockDim.x + threadIdx.x) >> 5;
    int lane = threadIdx.x & 31;
    if (wave >= MT * NT) return;
    int mt = wave / NT, nt = wave % NT;
    int row0 = mt * 16, n0 = nt * 16;
    int half = lane >> 4, m = lane & 15;
    int row = row0 + m, n = n0 + m;

    v8f acc = gemm2_tile(h, sW, row, n, half);

    if (n < F_OUT) {
        float bias = sb[n];
        #pragma unroll
        for (int i = 0; i < 8; ++i) {
            int mm = half * 8 + i;
            int r  = row0 + mm;
            float v = acc[i] + bias + agg2[(size_t)r * F_OUT + n] * invdeg[r];
            out[(size_t)r * F_OUT + n] = v;
        }
    }
}

// ---------------------------------------------------------------------------
// In-place row-wise log_softmax over 24 columns
// ---------------------------------------------------------------------------
__global__ void logsoftmax_kernel(float* __restrict__ out) {
    int r = blockIdx.x * blockDim.x + threadIdx.x;
    if (r >= N_NODES) return;
    float* p = out + (size_t)r * F_OUT;
    float v[F_OUT];
    float mx = -INFINITY;
    #pragma unroll
    for (int j = 0; j < F_OUT; ++j) { v[j] = p[j]; mx = fmaxf(mx, v[j]); }
    float s = 0.0f;
    #pragma unroll
    for (int j = 0; j < F_OUT; ++j) s += __expf(v[j] - mx);
    float l = __logf(s);
    #pragma unroll
    for (int j = 0; j < F_OUT; ++j) p[j] = v[j] - mx - l;
}

// ---------------------------------------------------------------------------
extern "C" void kernel_launch(void* const* d_in, const int* in_sizes, int n_in,
                              void* d_out, int out_size, void* d_ws, size_t ws_size,
                              hipStream_t stream) {
    const float*     x   = (const float*)d_in[0];
    const long long* ei  = (const long long*)d_in[1];   // int64 [2, N_EDGES]
    const float*     W1l = (const float*)d_in[2];
    const float*     b1  = (const float*)d_in[3];
    const float*     W1r = (const float*)d_in[4];
    const float*     W2l = (const float*)d_in[5];
    const float*     b2  = (const float*)d_in[6];
    const float*     W2r = (const float*)d_in[7];
    float*           out = (float*)d_out;

    const long long* src = ei;
    const long long* tgt = ei + N_EDGES;

    // ws layout (floats): deg[200k] | agg1[200k*16] | agg2[200k*24] | h[200k*40] | p2[200k*24]
    float* ws   = (float*)d_ws;
    float* deg  = ws;                              // becomes inv_deg in place
    float* agg1 = ws + (size_t)N_NODES;
    float* agg2 = ws + (size_t)N_NODES * 17;
    float* h    = ws + (size_t)N_NODES * 41;
    float* p2   = ws + (size_t)N_NODES * 81;

    // zero atomic targets: deg | agg1 | agg2 are contiguous [0, 41*N_NODES)
    hipMemsetAsync(ws, 0, (size_t)N_NODES * 41 * sizeof(float), stream);

    agg1_kernel<<<(N_EDGES * 4 + 255) / 256, 256, 0, stream>>>(x, src, tgt, agg1, deg);
    invdeg_kernel<<<(N_NODES + 255) / 256, 256, 0, stream>>>(deg);

    layer1_kernel<<<(MT * 3 + 7) / 8, 256, 0, stream>>>(agg1, x, deg, W1l, b1, W1r, h);

    gemm_p2_kernel<<<(MT * 2 + 7) / 8, 256, 0, stream>>>(h, W2l, p2);

    agg2_kernel<<<(N_EDGES * 6 + 255) / 256, 256, 0, stream>>>(p2, src, tgt, agg2);

    layer2_out_kernel<<<(MT * 2 + 7) / 8, 256, 0, stream>>>(h, W2r, agg2, deg, b2, out);

    logsoftmax_kernel<<<(N_NODES + 255) / 256, 256, 0, stream>>>(out);
}